// BAGCN_20194936226005
// MI455X (gfx1250) — compile-verified
//
#include <hip/hip_runtime.h>
#include <math.h>

#define D 200
#define EPS 1e-16f

typedef __attribute__((ext_vector_type(2))) float v2f;
typedef __attribute__((ext_vector_type(8))) float v8f;

// ---------------------------------------------------------------- zero fill
__global__ __launch_bounds__(256) void k_zero(float* __restrict__ p, size_t n) {
  size_t i = (size_t)blockIdx.x * blockDim.x + threadIdx.x;
  size_t stride = (size_t)gridDim.x * blockDim.x;
  for (; i < n; i += stride) p[i] = 0.f;
}

// ---------------------------------------------------------------- WMMA GEMM
// C[M,N] = A[M,K] (row-major, lda) * W[N,K]^T (W row-major, ldw)  (+= if acc)
// fp32 WMMA 16x16x4: A lanes0-15 hold M=r {K=k,k+1}, lanes16-31 {K=k+2,k+3};
// B lanes0-15 hold N=r {K=k,k+1}, lanes16-31 {K=k+2,k+3};
// C vgpr j: lanes0-15 -> (M=j, N=lane), lanes16-31 -> (M=j+8, N=lane-16).
__global__ __launch_bounds__(128) void k_gemm_wmma(
    const float* __restrict__ A, int lda,
    const float* __restrict__ W, int ldw,
    float* __restrict__ C, int ldc,
    int M, int N, int K, int accumulate)
{
  int mt = (M + 15) >> 4, nt = (N + 15) >> 4;
  int wave = blockIdx.x * (blockDim.x >> 5) + (threadIdx.x >> 5);
  if (wave >= mt * nt) return;              // wave-uniform: EXEC stays full
  int tm = wave / nt, tn = wave % nt;
  int lane = threadIdx.x & 31;
  int half = lane >> 4, r = lane & 15;
  int arow = tm * 16 + r; if (arow >= M) arow = M - 1;   // clamp, stores masked
  int brow = tn * 16 + r; if (brow >= N) brow = N - 1;
  const float* Ap = A + (size_t)arow * lda + 2 * half;
  const float* Wp = W + (size_t)brow * ldw + 2 * half;
  v8f acc = {};
#pragma unroll 2
  for (int k = 0; k < K; k += 4) {
    v2f av = *(const v2f*)(Ap + k);
    v2f bv = *(const v2f*)(Wp + k);
    acc = __builtin_amdgcn_wmma_f32_16x16x4_f32(
        false, av, false, bv, (short)0, acc, false, false);
  }
  int col = tn * 16 + r;
  if (col < N) {
#pragma unroll
    for (int j = 0; j < 8; ++j) {
      int row = tm * 16 + j + 8 * half;
      if (row < M) {
        size_t idx = (size_t)row * ldc + col;
        float v = acc[j];
        if (accumulate) v += C[idx];
        C[idx] = v;
      }
    }
  }
}

// ---------------------------------------------------------------- row dot a
__global__ __launch_bounds__(128) void k_rowdot(
    const float* __restrict__ X, const float* __restrict__ a,
    float* __restrict__ out, int rows)
{
  int row = blockIdx.x * (blockDim.x >> 5) + (threadIdx.x >> 5);
  if (row >= rows) return;
  int lane = threadIdx.x & 31;
  float s = 0.f;
  for (int d = lane; d < D; d += 32) s += X[(size_t)row * D + d] * a[d];
  for (int off = 16; off; off >>= 1) s += __shfl_down(s, off, 32);
  if (lane == 0) out[row] = s;
}

// ---------------------------------------------------------------- edge scores
__device__ __forceinline__ unsigned f2key(float f) {
  unsigned b = __float_as_uint(f);
  return (b & 0x80000000u) ? ~b : (b | 0x80000000u);
}
__device__ __forceinline__ float key2f(unsigned k) {
  return (k & 0x80000000u) ? __uint_as_float(k ^ 0x80000000u)
                           : __uint_as_float(~k);
}

__global__ __launch_bounds__(256) void k_edge_scores(
    const int* __restrict__ subs, const int* __restrict__ objs,
    const int* __restrict__ typs,
    const float* __restrict__ pa_sub, const float* __restrict__ pa_obj,
    const float* __restrict__ pa_rel,
    float* __restrict__ scores, unsigned* __restrict__ gmax, int E)
{
  int e = blockIdx.x * blockDim.x + threadIdx.x;
  float s = -3.4e38f;
  if (e < E) {
    float v = pa_sub[subs[e]] + pa_obj[objs[e]] + pa_rel[typs[e]];
    s = (v > 0.f) ? v : 0.2f * v;          // leaky_relu slope 0.2
    scores[e] = s;
  }
  for (int off = 16; off; off >>= 1) s = fmaxf(s, __shfl_down(s, off, 32));
  if ((threadIdx.x & 31) == 0) atomicMax(gmax, f2key(s));
}

// ------------------------------------------------- exp + segment denominators
__global__ __launch_bounds__(256) void k_edge_expdenom(
    const int* __restrict__ subs, const int* __restrict__ typs, int toff,
    float* __restrict__ scores, const unsigned* __restrict__ gmax,
    float* __restrict__ denom_ent, float* __restrict__ denom_rel, int E, int NR)
{
  __shared__ float srel[512];              // NR <= 512 (actual 200)
  for (int i = threadIdx.x; i < NR; i += blockDim.x) srel[i] = 0.f;
  __syncthreads();
  float mx = key2f(*gmax);
  int e = blockIdx.x * blockDim.x + threadIdx.x;
  if (e < E) {
    float ex = expf(scores[e] - mx);
    scores[e] = ex;                        // reuse buffer: now holds exp_s
    atomicAdd(&denom_ent[subs[e]], ex);
    atomicAdd(&srel[typs[e] - toff], ex);  // LDS float atomic (ds_add_f32)
  }
  __syncthreads();
  for (int i = threadIdx.x; i < NR; i += blockDim.x) {
    float v = srel[i];
    if (v != 0.f) atomicAdd(&denom_rel[i], v);
  }
}

// ---------------------------------------------------------------- aggregation
// triple = sub_proj[s] + obj_proj[o] + rel_proj[t]
// agg_ent[s] += triple*att_e (global f32 atomics, L2 resident)
// agg_rel[t-toff] += triple*att_r (accumulated in 160KB LDS tile, one flush/blk)
#define AGG_EDGES 2048
__global__ __launch_bounds__(256) void k_edge_aggregate(
    const int* __restrict__ subs, const int* __restrict__ objs,
    const int* __restrict__ typs, int toff,
    const float* __restrict__ sub_proj, const float* __restrict__ obj_proj,
    const float* __restrict__ rel_proj, const float* __restrict__ expS,
    const float* __restrict__ denom_ent, const float* __restrict__ denom_rel,
    float* __restrict__ agg_ent, float* __restrict__ agg_rel, int E, int NR)
{
  extern __shared__ float lrel[];          // NR*D floats = 160000 B (< 320KB WGP LDS)
  int tot = NR * D;
  for (int i = threadIdx.x; i < tot; i += blockDim.x) lrel[i] = 0.f;
  __syncthreads();
  int e0 = blockIdx.x * AGG_EDGES;
  int e1 = e0 + AGG_EDGES; if (e1 > E) e1 = E;
  int wid = threadIdx.x >> 5, lane = threadIdx.x & 31, nw = blockDim.x >> 5;
  for (int e = e0 + wid; e < e1; e += nw) {
    int s = subs[e], o = objs[e], t = typs[e];
    int tr = t - toff;
    float ex = expS[e];
    float we = ex / (denom_ent[s] + EPS);
    float wr = ex / (denom_rel[tr] + EPS);
    const float* sp = sub_proj + (size_t)s * D;
    const float* op = obj_proj + (size_t)o * D;
    const float* rp = rel_proj + (size_t)t * D;
    float* ae = agg_ent + (size_t)s * D;
    float* ar = lrel + tr * D;
    for (int d = lane; d < D; d += 32) {
      float tv = sp[d] + op[d] + rp[d];
      atomicAdd(&ae[d], tv * we);
      atomicAdd(&ar[d], tv * wr);          // LDS atomic
    }
  }
  __syncthreads();
  for (int i = threadIdx.x; i < tot; i += blockDim.x) {
    float v = lrel[i];
    if (v != 0.f) atomicAdd(&agg_rel[i], v);
  }
}

// ---------------------------------------------------------------- gate fuse
// out = (1-sig(G2+gbias))*xent + sig(G2+gbias)*tanh(G1+gb) + bias
__global__ __launch_bounds__(256) void k_gate(
    const float* __restrict__ xent, const float* __restrict__ glin,
    const float* __restrict__ gatelin, const float* __restrict__ gb,
    const float* __restrict__ gbias, const float* __restrict__ bias,
    float* __restrict__ out, size_t total)
{
  size_t i = (size_t)blockIdx.x * blockDim.x + threadIdx.x;
  if (i >= total) return;
  int d = (int)(i % D);
  float g = 1.f / (1.f + expf(-(gatelin[i] + gbias[d])));
  float ge = tanhf(glin[i] + gb[d]);
  out[i] = (1.f - g) * xent[i] + g * ge + bias[d];
}

// ---------------------------------------------------------------- host side
static inline void gemm(const float* A, int lda, const float* W, int ldw,
                        float* C, int ldc, int M, int N, int K, int acc,
                        hipStream_t s)
{
  int mt = (M + 15) / 16, nt = (N + 15) / 16;
  int waves = mt * nt;
  int blocks = (waves + 3) / 4;
  k_gemm_wmma<<<blocks, 128, 0, s>>>(A, lda, W, ldw, C, ldc, M, N, K, acc);
}

extern "C" void kernel_launch(void* const* d_in, const int* in_sizes, int n_in,
                              void* d_out, int out_size, void* d_ws, size_t ws_size,
                              hipStream_t stream)
{
  const float* ent      = (const float*)d_in[0];
  const float* rel      = (const float*)d_in[1];
  const int*   eidx     = (const int*)d_in[2];
  const int*   etyp     = (const int*)d_in[3];
  const float* w_sub    = (const float*)d_in[4];
  const float* w_obj    = (const float*)d_in[5];
  const float* w_rel    = (const float*)d_in[6];
  const float* avec     = (const float*)d_in[7];
  const float* k_in     = (const float*)d_in[8];
  const float* k_out    = (const float*)d_in[9];
  const float* k_rel    = (const float*)d_in[10];
  const float* resW_ent = (const float*)d_in[11];
  const float* resW_rel = (const float*)d_in[12];
  const float* bias_ent = (const float*)d_in[13];
  const float* bias_rel = (const float*)d_in[14];
  const float* ge_gW    = (const float*)d_in[15]; // (D, 3D)
  const float* ge_gb    = (const float*)d_in[16];
  const float* ge_g1W   = (const float*)d_in[17]; // (D, D)
  const float* ge_g2W   = (const float*)d_in[18]; // (D, 2D)
  const float* ge_gbias = (const float*)d_in[19];
  const float* gr_gW    = (const float*)d_in[20]; // (D, 2D)
  const float* gr_gb    = (const float*)d_in[21];
  const float* gr_g1W   = (const float*)d_in[22]; // (D, D)
  const float* gr_g2W   = (const float*)d_in[23]; // (D, D)
  const float* gr_gbias = (const float*)d_in[24];

  const int NE  = in_sizes[0] / D;   // 50000
  const int NR2 = in_sizes[1] / D;   // 400
  const int NR  = NR2 / 2;           // 200
  const int E2  = in_sizes[3];       // 1,000,000
  const int E   = E2 / 2;            // 500,000

  float* ws = (float*)d_ws;
  // workspace layout (floats); total ~41.7M floats (~167 MB)
  size_t f_sub     = 0;                               // sub_proj -> later in_ent
  size_t f_obj     = f_sub     + (size_t)NE * D;      // obj_proj -> later out_ent
  size_t f_relproj = f_obj     + (size_t)NE * D;      // rel_proj (NR2 x D)
  size_t f_pasub   = f_relproj + (size_t)NR2 * D;
  size_t f_paobj   = f_pasub   + NE;
  size_t f_parel   = f_paobj   + NE;
  size_t f_scores  = f_parel   + NR2;                 // E2 (scores -> exp_s)
  size_t f_resrel  = f_scores  + (size_t)E2;
  size_t f_catrel  = f_resrel  + (size_t)NR2 * D;     // update_rel pre-gate
  size_t f_H1      = f_catrel  + (size_t)NR2 * D;
  size_t f_H2      = f_H1      + (size_t)NR2 * D;
  size_t f_zero    = f_H2      + (size_t)NR2 * D;     // start of zeroed region
  size_t f_gmax    = f_zero;                          // 2 uints
  size_t f_dent    = f_gmax    + 2;                   // 2*NE
  size_t f_drel    = f_dent    + 2 * (size_t)NE;      // 2*NR
  size_t f_aggin   = f_drel    + 2 * (size_t)NR;      // NE x D -> later res_ent
  size_t f_aggout  = f_aggin   + (size_t)NE * D;      // NE x D -> later G1
  size_t f_aggrel  = f_aggout  + (size_t)NE * D;      // NR2 x D (both dirs)
  size_t f_end     = f_aggrel  + (size_t)NR2 * D;

  // 1) zero all accumulated buffers (atomics target them)
  k_zero<<<2048, 256, 0, stream>>>(ws + f_zero, f_end - f_zero);

  // 2) dense projections (fp32 WMMA)
  gemm(ent, D, w_sub, D, ws + f_sub,     D, NE,  D, D, 0, stream);
  gemm(ent, D, w_obj, D, ws + f_obj,     D, NE,  D, D, 0, stream);
  gemm(rel, D, w_rel, D, ws + f_relproj, D, NR2, D, D, 0, stream);

  // 3) precompute dot(row, a) so edge scores are 3 scalar gathers each
  k_rowdot<<<(NE  + 3) / 4, 128, 0, stream>>>(ws + f_sub,     avec, ws + f_pasub, NE);
  k_rowdot<<<(NE  + 3) / 4, 128, 0, stream>>>(ws + f_obj,     avec, ws + f_paobj, NE);
  k_rowdot<<<(NR2 + 3) / 4, 128, 0, stream>>>(ws + f_relproj, avec, ws + f_parel, NR2);

  // 4) per-direction edge phase
  for (int dir = 0; dir < 2; ++dir) {
    const int* subs = eidx + (size_t)dir * E;
    const int* objs = eidx + (size_t)E2 + (size_t)dir * E;
    const int* typs = etyp + (size_t)dir * E;
    float* scores   = ws + f_scores + (size_t)dir * E;
    unsigned* gmax  = (unsigned*)(ws + f_gmax) + dir;
    float* dent     = ws + f_dent + (size_t)dir * NE;
    float* drel     = ws + f_drel + (size_t)dir * NR;
    float* aggent   = ws + ((dir == 0) ? f_aggin : f_aggout);
    float* aggrel   = ws + f_aggrel + (size_t)dir * NR * D;
    int toff = dir * NR;
    int blocks = (E + 255) / 256;
    k_edge_scores<<<blocks, 256, 0, stream>>>(subs, objs, typs,
        ws + f_pasub, ws + f_paobj, ws + f_parel, scores, gmax, E);
    k_edge_expdenom<<<blocks, 256, 0, stream>>>(subs, typs, toff,
        scores, gmax, dent, drel, E, NR);
    int ablocks = (E + AGG_EDGES - 1) / AGG_EDGES;
    size_t lds = (size_t)NR * D * sizeof(float);   // 160 KB rel tile
    k_edge_aggregate<<<ablocks, 256, lds, stream>>>(subs, objs, typs, toff,
        ws + f_sub, ws + f_obj, ws + f_relproj, scores, dent, drel,
        aggent, aggrel, E, NR);
  }

  // 5) output kernels: in_ent/out_ent overwrite the proj buffers (now free)
  gemm(ws + f_aggin,  D, k_in,  D, ws + f_sub, D, NE, D, D, 0, stream);
  gemm(ws + f_aggout, D, k_out, D, ws + f_obj, D, NE, D, D, 0, stream);
  gemm(ws + f_aggrel,                 D, k_rel, D, ws + f_catrel,                 D, NR, D, D, 0, stream);
  gemm(ws + f_aggrel + (size_t)NR*D,  D, k_rel, D, ws + f_catrel + (size_t)NR*D,  D, NR, D, D, 0, stream);

  // 6) residuals (res_ent reuses agg_in buffer, now free)
  gemm(ent, D, resW_ent, D, ws + f_aggin,  D, NE,  D, D, 0, stream);
  gemm(rel, D, resW_rel, D, ws + f_resrel, D, NR2, D, D, 0, stream);

  // 7) entity gate: x = [res_ent | in_ent | out_ent]; decompose concat-GEMMs
  float* G1 = ws + f_aggout;       // g_emb linear (agg_out buffer now free)
  float* G2 = (float*)d_out;       // gate linear staged in d_out
  gemm(ws + f_aggin, D, ge_gW + 0,     3 * D, G1, D, NE, D, D, 0, stream);
  gemm(ws + f_sub,   D, ge_gW + D,     3 * D, G1, D, NE, D, D, 1, stream);
  gemm(ws + f_obj,   D, ge_gW + 2 * D, 3 * D, G1, D, NE, D, D, 1, stream);
  gemm(ws + f_aggin, D, ge_g1W,        D,     G2, D, NE, D, D, 0, stream);
  gemm(ws + f_sub,   D, ge_g2W + 0,    2 * D, G2, D, NE, D, D, 1, stream);
  gemm(ws + f_obj,   D, ge_g2W + D,    2 * D, G2, D, NE, D, D, 1, stream);
  {
    size_t total = (size_t)NE * D;
    k_gate<<<(int)((total + 255) / 256), 256, 0, stream>>>(
        ws + f_aggin, G1, G2, ge_gb, ge_gbias, bias_ent, (float*)d_out, total);
  }

  // 8) relation gate: x_ent = cat_rel (in_rel;out_rel), x_lit = res_rel
  gemm(ws + f_catrel, D, gr_gW + 0, 2 * D, ws + f_H1, D, NR2, D, D, 0, stream);
  gemm(ws + f_resrel, D, gr_gW + D, 2 * D, ws + f_H1, D, NR2, D, D, 1, stream);
  gemm(ws + f_catrel, D, gr_g1W,    D,     ws + f_H2, D, NR2, D, D, 0, stream);
  gemm(ws + f_resrel, D, gr_g2W,    D,     ws + f_H2, D, NR2, D, D, 1, stream);
  {
    size_t total = (size_t)NR2 * D;
    k_gate<<<(int)((total + 255) / 256), 256, 0, stream>>>(
        ws + f_catrel, ws + f_H1, ws + f_H2, gr_gb, gr_gbias, bias_rel,
        (float*)d_out + (size_t)NE * D, total);
  }
}